// MixGAT_48576080118128
// MI455X (gfx1250) — compile-verified
//
#include <hip/hip_runtime.h>
#include <hip/hip_bf16.h>

// ---------------------------------------------------------------------------
// MixGAT (2x GATv2 + FC) for gfx1250.
// Dense GEMMs: V_WMMA_F32_16X16X4_F32, 5 M-tiles per wave (B reused 5x/load),
// NO tail guards (M/16 == 625 == 125*5) so loops are fully uniform and the
// compiler can cluster loads and issue WMMAs back-to-back.
// Edge work: wave-per-edge float4 gathers from L2-resident xl/xr; segment
// softmax via int-punned atomicMax + fp32 global atomic adds.
// ---------------------------------------------------------------------------

typedef __attribute__((ext_vector_type(2))) float v2f;
typedef __attribute__((ext_vector_type(8))) float v8f;

#if defined(__gfx1250__)
#if __has_builtin(__builtin_amdgcn_wmma_f32_16x16x4_f32)
#define HAVE_WMMA_F32 1
#endif
#endif
#ifndef HAVE_WMMA_F32
#define HAVE_WMMA_F32 0
#endif

#define MT 5                            // M-tiles per wave; M/16 must be % MT
#define ORD_NEG_INF ((int)0x807FFFFF)   // f2ord(-inf)

__device__ __forceinline__ int f2ord(float f) {
  int i = __float_as_int(f);
  return i >= 0 ? i : (i ^ 0x7FFFFFFF);
}
__device__ __forceinline__ float ord2f(int k) {
  return __int_as_float(k >= 0 ? k : (k ^ 0x7FFFFFFF));
}

// -------------------------------- init -------------------------------------

__global__ void init_bias_kernel(float* __restrict__ p, const float* __restrict__ b,
                                 int total, int periodMask) {
  int i = blockIdx.x * blockDim.x + threadIdx.x;
  if (i < total) p[i] = b[i & periodMask];
}

__global__ void init_i32_kernel(int* __restrict__ p, int v, int n) {
  int i = blockIdx.x * blockDim.x + threadIdx.x;
  if (i < n) p[i] = v;
}

__global__ void init_f32_kernel(float* __restrict__ p, float v, int n) {
  int i = blockIdx.x * blockDim.x + threadIdx.x;
  if (i < n) p[i] = v;
}

// ------------------------------- GEMM --------------------------------------
// C[M,N] = A[M,K] @ B[K,N] + bias[N]; row-major; (M/16)%MT==0, N%16==0, K%4==0.
// One wave per (MT*16) x 16 C block: one B pair per K-step feeds MT WMMAs.

__global__ void gemm_wmma_kernel(const float* __restrict__ A, const float* __restrict__ B,
                                 const float* __restrict__ bias, float* __restrict__ C,
                                 int M, int K, int N) {
  const int lane = threadIdx.x & 31;
  const int wave = (int)((blockIdx.x * blockDim.x + threadIdx.x) >> 5);
  const int ntiles = N >> 4;
  const int mgroups = (M >> 4) / MT;
  if (wave >= mgroups * ntiles) return;
  const int mg = wave / ntiles, nt = wave % ntiles;
  const int mt0 = mg * MT;
  const int row = lane & 15;
  const int n = (nt << 4) + row;
  const int hi = lane >> 4;  // 0 or 1
#if HAVE_WMMA_F32
  const float* arow[MT];
#pragma unroll
  for (int t = 0; t < MT; ++t)
    arow[t] = A + (size_t)((mt0 + t) * 16 + row) * K;
  v8f acc[MT];
#pragma unroll
  for (int t = 0; t < MT; ++t) acc[t] = (v8f){};
#pragma unroll 2
  for (int k0 = 0; k0 < K; k0 += 4) {
    const int k = k0 + 2 * hi;
    v2f b;
    b.x = B[(size_t)k * N + n];
    b.y = B[(size_t)(k + 1) * N + n];
    v2f a[MT];
#pragma unroll
    for (int t = 0; t < MT; ++t) a[t] = *(const v2f*)(arow[t] + k);
#pragma unroll
    for (int t = 0; t < MT; ++t)
      acc[t] = __builtin_amdgcn_wmma_f32_16x16x4_f32(false, a[t], false, b, (short)0,
                                                     acc[t], false, false);
  }
  const float bv = bias[n];
#pragma unroll
  for (int t = 0; t < MT; ++t) {
    const int mbase = (mt0 + t) << 4;
#pragma unroll
    for (int v = 0; v < 8; ++v)
      C[(size_t)(mbase + v + 8 * hi) * N + n] = acc[t][v] + bv;
  }
#else
  const float bv = bias[n];
  for (int t = 0; t < MT; ++t) {
    const int mbase = (mt0 + t) << 4;
    for (int v = 0; v < 8; ++v) {
      const int m = mbase + v + 8 * hi;
      float s = 0.f;
      for (int k = 0; k < K; ++k) s += A[(size_t)m * K + k] * B[(size_t)k * N + n];
      C[(size_t)m * N + n] = s + bv;
    }
  }
#endif
}

// Final GEMM reading a virtual concat [A0 (M x K0) | A1 (M x K1)] @ B[K0+K1, N].
__global__ void gemm_concat_wmma_kernel(const float* __restrict__ A0,
                                        const float* __restrict__ A1,
                                        const float* __restrict__ B,
                                        const float* __restrict__ bias,
                                        float* __restrict__ C,
                                        int M, int K0, int K1, int N) {
  const int lane = threadIdx.x & 31;
  const int wave = (int)((blockIdx.x * blockDim.x + threadIdx.x) >> 5);
  const int ntiles = N >> 4;
  const int mgroups = (M >> 4) / MT;
  if (wave >= mgroups * ntiles) return;
  const int mg = wave / ntiles, nt = wave % ntiles;
  const int mt0 = mg * MT;
  const int row = lane & 15;
  const int n = (nt << 4) + row;
  const int hi = lane >> 4;
  const int K = K0 + K1;
#if HAVE_WMMA_F32
  const float* p0[MT];
  const float* p1[MT];
#pragma unroll
  for (int t = 0; t < MT; ++t) {
    const int m = (mt0 + t) * 16 + row;
    p0[t] = A0 + (size_t)m * K0;
    p1[t] = A1 + (size_t)m * K1 - K0;  // so that p1[t] + k == A1[m, k-K0]
  }
  v8f acc[MT];
#pragma unroll
  for (int t = 0; t < MT; ++t) acc[t] = (v8f){};
#pragma unroll 2
  for (int k0 = 0; k0 < K; k0 += 4) {
    const int k = k0 + 2 * hi;
    v2f b;
    b.x = B[(size_t)k * N + n];
    b.y = B[(size_t)(k + 1) * N + n];
    const bool second = (k0 >= K0);  // pairs never straddle (K0 % 4 == 0)
    v2f a[MT];
#pragma unroll
    for (int t = 0; t < MT; ++t) a[t] = *(const v2f*)((second ? p1[t] : p0[t]) + k);
#pragma unroll
    for (int t = 0; t < MT; ++t)
      acc[t] = __builtin_amdgcn_wmma_f32_16x16x4_f32(false, a[t], false, b, (short)0,
                                                     acc[t], false, false);
  }
  const float bv = bias[n];
#pragma unroll
  for (int t = 0; t < MT; ++t) {
    const int mbase = (mt0 + t) << 4;
#pragma unroll
    for (int v = 0; v < 8; ++v)
      C[(size_t)(mbase + v + 8 * hi) * N + n] = acc[t][v] + bv;
  }
#else
  const float bv = bias[n];
  for (int t = 0; t < MT; ++t) {
    const int mbase = (mt0 + t) << 4;
    for (int v = 0; v < 8; ++v) {
      const int m = mbase + v + 8 * hi;
      float s = 0.f;
      for (int k = 0; k < K; ++k) {
        float av = (k < K0) ? A0[(size_t)m * K0 + k] : A1[(size_t)m * K1 + (k - K0)];
        s += av * B[(size_t)k * N + n];
      }
      C[(size_t)m * N + n] = s + bv;
    }
  }
#endif
}

// ----------------------------- edge passes ---------------------------------
// Pass A: one wave per edge; each lane owns 4 consecutive channels (b128 loads).
__global__ void edge_logits_kernel(const float* __restrict__ xl, const float* __restrict__ xr,
                                   const float* __restrict__ att,
                                   const int* __restrict__ srcA, const int* __restrict__ dstA,
                                   int E, int Nn, int H,
                                   float* __restrict__ logits, int* __restrict__ mmax) {
  const int wave = (int)((blockIdx.x * blockDim.x + threadIdx.x) >> 5);
  const int lane = threadIdx.x & 31;
  const int Etot = E + Nn;
  if (wave >= Etot) return;
  const int src = (wave < E) ? srcA[wave] : (wave - E);
  const int dst = (wave < E) ? dstA[wave] : (wave - E);
  const int HC = H << 7;
  const int c4 = lane << 2;  // 0..124, one float4 per lane
  const float* pl = xl + (size_t)src * HC + c4;
  const float* pr = xr + (size_t)dst * HC + c4;
  const float* pa = att + c4;
  for (int h = 0; h < H; ++h) {
    const float4 l = *(const float4*)(pl + h * 128);
    const float4 r = *(const float4*)(pr + h * 128);
    const float4 a = *(const float4*)(pa + h * 128);
    float s0 = l.x + r.x; s0 = (s0 > 0.f) ? s0 : 0.2f * s0;
    float s1 = l.y + r.y; s1 = (s1 > 0.f) ? s1 : 0.2f * s1;
    float s2 = l.z + r.z; s2 = (s2 > 0.f) ? s2 : 0.2f * s2;
    float s3 = l.w + r.w; s3 = (s3 > 0.f) ? s3 : 0.2f * s3;
    float sum = s0 * a.x + s1 * a.y + s2 * a.z + s3 * a.w;
    for (int off = 16; off > 0; off >>= 1) sum += __shfl_xor(sum, off, 32);
    if (lane == 0) {
      logits[(size_t)wave * H + h] = sum;
      atomicMax(&mmax[dst * H + h], f2ord(sum));
    }
  }
}

// Pass B: one thread per (edge, head): a = exp(logit - m[dst]); denom += a.
__global__ void edge_exp_kernel(const int* __restrict__ dstA, int E, int Nn, int H,
                                float* __restrict__ logits, const int* __restrict__ mmax,
                                float* __restrict__ denom) {
  const int tid = blockIdx.x * blockDim.x + threadIdx.x;
  const int total = (E + Nn) * H;
  if (tid >= total) return;
  const int e = tid / H, h = tid - e * H;
  const int dst = (e < E) ? dstA[e] : (e - E);
  const float m = ord2f(mmax[dst * H + h]);
  const float a = __expf(logits[tid] - m);
  logits[tid] = a;
  unsafeAtomicAdd(&denom[dst * H + h], a);
}

// Pass C: one wave per edge; b128 gather of xl[src], fp32 atomic scatter-add.
__global__ void edge_accum_kernel(const float* __restrict__ xl, const float* __restrict__ logits,
                                  const float* __restrict__ denom,
                                  const int* __restrict__ srcA, const int* __restrict__ dstA,
                                  int E, int Nn, int H, float* __restrict__ out) {
  const int wave = (int)((blockIdx.x * blockDim.x + threadIdx.x) >> 5);
  const int lane = threadIdx.x & 31;
  const int Etot = E + Nn;
  if (wave >= Etot) return;
  const int src = (wave < E) ? srcA[wave] : (wave - E);
  const int dst = (wave < E) ? dstA[wave] : (wave - E);
  const int HC = H << 7;
  const int c4 = lane << 2;
  const float* pl = xl + (size_t)src * HC + c4;
  float* po = out + (size_t)dst * HC + c4;
  for (int h = 0; h < H; ++h) {
    const float alpha = logits[(size_t)wave * H + h] / denom[dst * H + h];
    const float4 l = *(const float4*)(pl + h * 128);
    float* q = po + h * 128;
    unsafeAtomicAdd(q + 0, alpha * l.x);
    unsafeAtomicAdd(q + 1, alpha * l.y);
    unsafeAtomicAdd(q + 2, alpha * l.z);
    unsafeAtomicAdd(q + 3, alpha * l.w);
  }
}

// ------------------------------- launch ------------------------------------

static inline int cdiv(long long a, long long b) { return (int)((a + b - 1) / b); }

static inline void launch_gemm(const float* A, const float* W, const float* bias, float* C,
                               int M, int K, int N, hipStream_t s) {
  long long mgroups = (M / 16) / MT;  // exact: 625 / 5 == 125
  long long waves = mgroups * (N / 16);
  int threads = 128;
  int blocks = cdiv(waves * 32, threads);
  gemm_wmma_kernel<<<blocks, threads, 0, s>>>(A, W, bias, C, M, K, N);
}

extern "C" void kernel_launch(void* const* d_in, const int* in_sizes, int n_in,
                              void* d_out, int out_size, void* d_ws, size_t ws_size,
                              hipStream_t stream) {
  const float* x    = (const float*)d_in[0];
  const int*   e1   = (const int*)d_in[1];
  const int*   e2   = (const int*)d_in[2];
  const float* Wl1  = (const float*)d_in[3];
  const float* bl1  = (const float*)d_in[4];
  const float* Wr1  = (const float*)d_in[5];
  const float* br1  = (const float*)d_in[6];
  const float* att1 = (const float*)d_in[7];
  const float* b1   = (const float*)d_in[8];
  const float* Wl2  = (const float*)d_in[9];
  const float* bl2  = (const float*)d_in[10];
  const float* Wr2  = (const float*)d_in[11];
  const float* br2  = (const float*)d_in[12];
  const float* att2 = (const float*)d_in[13];
  const float* b2   = (const float*)d_in[14];
  const float* Wfc  = (const float*)d_in[15];
  const float* bfc  = (const float*)d_in[16];

  const int d  = 128;
  const int N  = in_sizes[0] / d;       // 10000
  const int E1 = in_sizes[1] / 2;       // 320000
  const int E2 = in_sizes[2] / 2;       // 640000
  const int H1 = 4, H2 = 2;
  const int HC1 = H1 * d;               // 512
  const int HC2 = H2 * d;               // 256
  const int E1t = E1 + N;
  const int E2t = E2 + N;

  const int* e1src = e1;
  const int* e1dst = e1 + E1;
  const int* e2src = e2;
  const int* e2dst = e2 + E2;

  // workspace carve-up (all offsets multiples of 4 floats -> 16B aligned)
  float* ws = (float*)d_ws;
  size_t off = 0;
  float* xl1  = ws + off; off += (size_t)N * HC1;
  float* xr1  = ws + off; off += (size_t)N * HC1;
  float* xl2  = ws + off; off += (size_t)N * HC2;
  float* xr2  = ws + off; off += (size_t)N * HC2;
  float* out1 = ws + off; off += (size_t)N * HC1;
  float* out2 = ws + off; off += (size_t)N * HC2;
  float* lg1  = ws + off; off += (size_t)E1t * H1;
  float* lg2  = ws + off; off += (size_t)E2t * H2;
  int*   m1   = (int*)(ws + off); off += (size_t)N * H1;
  float* den1 = ws + off; off += (size_t)N * H1;
  int*   m2   = (int*)(ws + off); off += (size_t)N * H2;
  float* den2 = ws + off; off += (size_t)N * H2;
  (void)ws_size; (void)n_in; (void)out_size;

  const int T = 256;

  // 1) init accumulators (seed with output biases), segment max, denominators
  init_bias_kernel<<<cdiv((size_t)N * HC1, T), T, 0, stream>>>(out1, b1, N * HC1, HC1 - 1);
  init_bias_kernel<<<cdiv((size_t)N * HC2, T), T, 0, stream>>>(out2, b2, N * HC2, HC2 - 1);
  init_i32_kernel<<<cdiv(N * H1, T), T, 0, stream>>>(m1, ORD_NEG_INF, N * H1);
  init_i32_kernel<<<cdiv(N * H2, T), T, 0, stream>>>(m2, ORD_NEG_INF, N * H2);
  init_f32_kernel<<<cdiv(N * H1, T), T, 0, stream>>>(den1, 0.f, N * H1);
  init_f32_kernel<<<cdiv(N * H2, T), T, 0, stream>>>(den2, 0.f, N * H2);

  // 2) dense transforms (fp32 WMMA, 5 M-tiles per wave, no tail)
  launch_gemm(x, Wl1, bl1, xl1, N, d, HC1, stream);
  launch_gemm(x, Wr1, br1, xr1, N, d, HC1, stream);
  launch_gemm(x, Wl2, bl2, xl2, N, d, HC2, stream);
  launch_gemm(x, Wr2, br2, xr2, N, d, HC2, stream);

  // 3) edge logits + segment max (wave per edge, b128 gathers)
  edge_logits_kernel<<<cdiv((long long)E1t * 32, T), T, 0, stream>>>(
      xl1, xr1, att1, e1src, e1dst, E1, N, H1, lg1, m1);
  edge_logits_kernel<<<cdiv((long long)E2t * 32, T), T, 0, stream>>>(
      xl2, xr2, att2, e2src, e2dst, E2, N, H2, lg2, m2);

  // 4) exp + denominator
  edge_exp_kernel<<<cdiv((long long)E1t * H1, T), T, 0, stream>>>(
      e1dst, E1, N, H1, lg1, m1, den1);
  edge_exp_kernel<<<cdiv((long long)E2t * H2, T), T, 0, stream>>>(
      e2dst, E2, N, H2, lg2, m2, den2);

  // 5) weighted scatter-add into out1/out2 (bias already seeded)
  edge_accum_kernel<<<cdiv((long long)E1t * 32, T), T, 0, stream>>>(
      xl1, lg1, den1, e1src, e1dst, E1, N, H1, out1);
  edge_accum_kernel<<<cdiv((long long)E2t * 32, T), T, 0, stream>>>(
      xl2, lg2, den2, e2src, e2dst, E2, N, H2, out2);

  // 6) final FC on the virtual concat [out1 | out2] (fp32 WMMA, K = 768)
  {
    long long mgroups = (N / 16) / MT;
    long long waves = mgroups * (d / 16);
    int blocks = cdiv(waves * 32, 128);
    gemm_concat_wmma_kernel<<<blocks, 128, 0, stream>>>(
        out1, out2, Wfc, bfc, (float*)d_out, N, HC1, HC2, d);
  }
}